// GaussianSplatGateUpInit_74191265071609
// MI455X (gfx1250) — compile-verified
//
#include <hip/hip_runtime.h>
#include <hip/hip_bf16.h>

typedef float v2f __attribute__((ext_vector_type(2)));
typedef float v8f __attribute__((ext_vector_type(8)));

#define BB_   2
#define KP_   1024
#define MM_   8
#define CC_   128
#define RR_   4
#define KC_   (KP_*MM_)          // 8192
#define NPAR  (BB_*KP_)          // 2048
#define NCH   (BB_*KC_)          // 16384

#define EPS_     1e-6f
#define JITTER_  1e-4f
#define INVPHI2  (1.0f/(1.6f*1.6f))
#define LOG2PI_  1.8378770664093453f

// d_out layout (floats): s_child0, mu_child, Sigma_child, g, Bmat
#define O_MU   ((size_t)NCH*CC_)        // 2097152
#define O_SIG  (O_MU  + (size_t)NCH*3)  // 2146304
#define O_G    (O_SIG + (size_t)NCH*9)  // 2293760
#define O_B    (O_G   + (size_t)NCH)    // 2310144

// workspace layout (float units)
#define W_L    0                        // [NPAR][6]  lower-tri chol
#define W_SINV (W_L    + NPAR*6)        // [NPAR][6]  sym inverse
#define W_LD   (W_SINV + NPAR*6)        // [NPAR]     logdet
#define W_XIW  (W_LD   + NPAR)          // [8][3]     child_emb@geo_w+geo_b
#define W_TOP  (W_XIW  + 32)            // [NPAR][4]  ints

// ---------------- kernel 1: per-parent Cholesky / inverse / logdet ----------------
__global__ __launch_bounds__(256) void prepK(const float* __restrict__ Sigma,
                                             float* __restrict__ wL,
                                             float* __restrict__ wSinv,
                                             float* __restrict__ wLd) {
  int pi = blockIdx.x * 256 + threadIdx.x;
  if (pi >= NPAR) return;
  const float* S = Sigma + (size_t)pi * 9;
  float s00 = S[0] + EPS_, s01 = S[1], s02 = S[2];
  float s10 = S[3], s11 = S[4] + EPS_, s12 = S[5];
  float s20 = S[6], s21 = S[7], s22 = S[8] + EPS_;
  // Cholesky (lower)
  float l00 = sqrtf(s00);
  float l10 = s10 / l00;
  float l20 = s20 / l00;
  float l11 = sqrtf(s11 - l10 * l10);
  float l21 = (s21 - l20 * l10) / l11;
  float l22 = sqrtf(s22 - l20 * l20 - l21 * l21);
  float* L = wL + pi * 6;
  L[0] = l00; L[1] = l10; L[2] = l11; L[3] = l20; L[4] = l21; L[5] = l22;
  wLd[pi] = 2.0f * (logf(l00) + logf(l11) + logf(l22));
  // symmetric inverse via adjugate
  float det = s00*(s11*s22 - s12*s21) - s01*(s10*s22 - s12*s20) + s02*(s10*s21 - s11*s20);
  float id = 1.0f / det;
  float* iv = wSinv + pi * 6;
  iv[0] = (s11*s22 - s12*s21) * id;   // i00
  iv[1] = (s02*s21 - s01*s22) * id;   // i01
  iv[2] = (s01*s12 - s02*s11) * id;   // i02
  iv[3] = (s00*s22 - s02*s20) * id;   // i11
  iv[4] = (s02*s10 - s00*s12) * id;   // i12
  iv[5] = (s00*s11 - s01*s10) * id;   // i22
}

// ---------------- kernel 2: xiw[m][a] = geo_b[a] + child_emb[m] . geo_w[:,a] ----------------
__global__ void geoK(const float* __restrict__ emb, const float* __restrict__ gw,
                     const float* __restrict__ gb, float* __restrict__ xiw) {
  int t = threadIdx.x;
  if (t < 24) {
    int m = t / 3, a = t % 3;
    float s = gb[a];
    for (int c = 0; c < CC_; ++c) s += emb[m * CC_ + c] * gw[c * 3 + a];
    xiw[t] = s;
  }
}

// ---------------- kernel 3: gate MLP via f32 WMMA -> g[B,Kc] ----------------
// block = 128 threads (4 waves), 64 rows; wave handles 16 rows x full 128 cols of h1.
// w1 chunk is stored k-pair interleaved so each B fragment is one ds_load_b64
// directly into an even-aligned VGPR pair (no repacking movs before v_wmma).
__global__ __launch_bounds__(128) void gateK(const float* __restrict__ sp,
                                             const float* __restrict__ emb,
                                             const float* __restrict__ mask,
                                             const float* __restrict__ lng,
                                             const float* __restrict__ lnb,
                                             const float* __restrict__ w1,
                                             const float* __restrict__ b1,
                                             const float* __restrict__ w2,
                                             const float* __restrict__ b2,
                                             float* __restrict__ out) {
  __shared__ float xs[64 * 132];    // normalized inputs, padded stride (even)
  __shared__ float w1p[16 * 288];   // k-pair interleaved w1 chunk: [kpair][n][2]
  const int tid  = threadIdx.x;
  const int lane = tid & 31;
  const int wave = tid >> 5;

  { // LayerNorm: 2 threads per row, 64 elems each
    int rl = tid >> 1, half = tid & 1;
    int gi = blockIdx.x * 64 + rl;
    int b = gi >> 13, ci = gi & (KC_ - 1), p = ci >> 3, m = ci & 7;
    const float* srow = sp + ((size_t)(b * KP_ + p)) * CC_;
    const float* erow = emb + m * CC_;
    float s1 = 0.f, s2 = 0.f;
    for (int j = 0; j < 64; ++j) {
      int c = half * 64 + j;
      float x = srow[c] + erow[c];
      xs[rl * 132 + c] = x;
      s1 += x; s2 += x * x;
    }
    s1 += __shfl_xor(s1, 1);
    s2 += __shfl_xor(s2, 1);
    float mean = s1 * (1.f / CC_);
    float var  = s2 * (1.f / CC_) - mean * mean;
    float rstd = rsqrtf(var + 1e-5f);
    for (int j = 0; j < 64; ++j) {
      int c = half * 64 + j;
      float x = xs[rl * 132 + c];
      xs[rl * 132 + c] = (x - mean) * rstd * lng[c] + lnb[c];
    }
  }

  v8f acc[8];
  #pragma unroll
  for (int t = 0; t < 8; ++t) acc[t] = (v8f){0.f,0.f,0.f,0.f,0.f,0.f,0.f,0.f};

  const int rowl = (wave << 4) + (lane & 15);
  const int ko   = (lane >> 4) << 1;   // 0 or 2 (A/B fragment K split)
  const int nn   = lane & 15;

  for (int kc = 0; kc < 4; ++kc) {     // K chunks of 32
    __syncthreads();
    // stage w1 rows [kc*32, kc*32+32) pair-interleaved: w1p[kp*288 + n*2 + j] = w1[kc*32+2kp+j][n]
    for (int t = tid; t < 16 * CC_; t += 128) {
      int kp = t >> 7, n = t & 127;
      const float* src = w1 + (size_t)(kc * 32 + kp * 2) * CC_ + n;
      v2f pv;
      pv.x = src[0];
      pv.y = src[CC_];
      *(v2f*)&w1p[kp * 288 + n * 2] = pv;
    }
    __syncthreads();
    #pragma unroll
    for (int kk = 0; kk < 8; ++kk) {   // sub-chunks of 4
      int kb = kk * 4 + ko;            // always even
      v2f a = *(const v2f*)&xs[rowl * 132 + kc * 32 + kb];
      #pragma unroll
      for (int ct = 0; ct < 8; ++ct) { // 8 col tiles of 16
        v2f bf = *(const v2f*)&w1p[(kb >> 1) * 288 + (ct * 16 + nn) * 2];
        acc[ct] = __builtin_amdgcn_wmma_f32_16x16x4_f32(
            false, a, false, bf, (short)0, acc[ct], false, false);
      }
    }
  }

  // epilogue: SiLU then dot with w2, reduce across the 16-lane half
  float part[8];
  #pragma unroll
  for (int v = 0; v < 8; ++v) part[v] = 0.f;
  #pragma unroll
  for (int ct = 0; ct < 8; ++ct) {
    int col = ct * 16 + nn;
    float b1v = b1[col], w2v = w2[col];
    #pragma unroll
    for (int v = 0; v < 8; ++v) {
      float y = acc[ct][v] + b1v;
      float h = y / (1.f + __expf(-y));   // silu
      part[v] += h * w2v;
    }
  }
  #pragma unroll
  for (int off = 8; off >= 1; off >>= 1) {
    #pragma unroll
    for (int v = 0; v < 8; ++v) part[v] += __shfl_xor(part[v], off);
  }
  if ((lane & 15) == 0) {
    float bias2 = b2[0];
    int rbase = blockIdx.x * 64 + wave * 16 + ((lane >> 4) << 3);
    #pragma unroll
    for (int v = 0; v < 8; ++v) {
      int gi = rbase + v;
      int bb = gi >> 13, ci = gi & (KC_ - 1), pp = ci >> 3;
      float bv = part[v] + bias2;
      float g = 1.f / (1.f + __expf(-bv));  // sigmoid (ETA=1)
      out[O_G + gi] = g * mask[bb * KP_ + pp];   // BETA=0: prior MLP has no effect
    }
  }
}

// ---------------- kernel 4: per-parent top-R by Gaussian overlap ----------------
__global__ __launch_bounds__(256) void topkK(const float* __restrict__ mu,
                                             const float* __restrict__ sinv,
                                             const float* __restrict__ ld,
                                             int* __restrict__ top) {
  __shared__ float cs[8 * 128];
  __shared__ int   cix[8 * 128];
  int wave = threadIdx.x >> 5, lane = threadIdx.x & 31;
  int pi = blockIdx.x * 8 + wave;
  int bb = pi >> 10, p = pi & (KP_ - 1);
  float mx = mu[(size_t)pi * 3 + 0], my = mu[(size_t)pi * 3 + 1], mz = mu[(size_t)pi * 3 + 2];
  float ts[4] = {-1e38f, -1e38f, -1e38f, -1e38f};
  int   ti[4] = {0, 0, 0, 0};
  for (int it = 0; it < KP_ / 32; ++it) {
    int k = it * 32 + lane;
    int kk = bb * KP_ + k;
    float dx = mx - mu[(size_t)kk * 3 + 0];
    float dy = my - mu[(size_t)kk * 3 + 1];
    float dz = mz - mu[(size_t)kk * 3 + 2];
    const float* iv = sinv + kk * 6;
    float maha = iv[0]*dx*dx + iv[3]*dy*dy + iv[5]*dz*dz
               + 2.f*(iv[1]*dx*dy + iv[2]*dx*dz + iv[4]*dy*dz);
    float sc = (k == p) ? 1e9f : -0.5f * (maha + ld[kk] + 3.f * LOG2PI_);
    if (sc > ts[3]) {
      if (sc > ts[0])      { ts[3]=ts[2];ti[3]=ti[2]; ts[2]=ts[1];ti[2]=ti[1]; ts[1]=ts[0];ti[1]=ti[0]; ts[0]=sc;ti[0]=k; }
      else if (sc > ts[1]) { ts[3]=ts[2];ti[3]=ti[2]; ts[2]=ts[1];ti[2]=ti[1]; ts[1]=sc;ti[1]=k; }
      else if (sc > ts[2]) { ts[3]=ts[2];ti[3]=ti[2]; ts[2]=sc;ti[2]=k; }
      else                 { ts[3]=sc;ti[3]=k; }
    }
  }
  int base = wave * 128 + lane * 4;
  #pragma unroll
  for (int j = 0; j < 4; ++j) { cs[base + j] = ts[j]; cix[base + j] = ti[j]; }
  __syncthreads();
  if (lane == 0) {
    int b0 = wave * 128;
    for (int r = 0; r < RR_; ++r) {
      float best = -1e38f; int bj = 0;
      for (int j = 0; j < 128; ++j) {
        float v = cs[b0 + j];
        if (v > best) { best = v; bj = j; }
      }
      top[pi * RR_ + r] = cix[b0 + bj];
      cs[b0 + bj] = -1e38f;
    }
  }
}

// ---------------- kernel 5: per-child softmax weights + all gathered outputs ----------------
__global__ __launch_bounds__(256) void childK(const float* __restrict__ sp,
                                              const float* __restrict__ mu,
                                              const float* __restrict__ Sigma,
                                              const float* __restrict__ xin,
                                              const float* __restrict__ wL,
                                              const float* __restrict__ wSinv,
                                              const float* __restrict__ wLd,
                                              const float* __restrict__ xiw,
                                              const int* __restrict__ top,
                                              float* __restrict__ out) {
  int wave = threadIdx.x >> 5, lane = threadIdx.x & 31;
  int gi = blockIdx.x * 8 + wave;
  int b = gi >> 13, ci = gi & (KC_ - 1), p = ci >> 3, m = ci & 7;
  int pi = b * KP_ + p;
  const float* Lp = wL + pi * 6;
  float xi0 = xin[((size_t)pi * MM_ + m) * 3 + 0] + xiw[m * 3 + 0];
  float xi1 = xin[((size_t)pi * MM_ + m) * 3 + 1] + xiw[m * 3 + 1];
  float xi2 = xin[((size_t)pi * MM_ + m) * 3 + 2] + xiw[m * 3 + 2];
  float mu0x = mu[(size_t)pi * 3 + 0] + Lp[0] * xi0;
  float mu0y = mu[(size_t)pi * 3 + 1] + Lp[1] * xi0 + Lp[2] * xi1;
  float mu0z = mu[(size_t)pi * 3 + 2] + Lp[3] * xi0 + Lp[4] * xi1 + Lp[5] * xi2;

  int kidx[4]; float ev[4]; float emax = -1e38f;
  #pragma unroll
  for (int r = 0; r < RR_; ++r) {
    int k = top[pi * RR_ + r]; kidx[r] = k;
    int kk = b * KP_ + k;
    float dx = mu0x - mu[(size_t)kk * 3 + 0];
    float dy = mu0y - mu[(size_t)kk * 3 + 1];
    float dz = mu0z - mu[(size_t)kk * 3 + 2];
    const float* iv = wSinv + kk * 6;
    float maha = iv[0]*dx*dx + iv[3]*dy*dy + iv[5]*dz*dz
               + 2.f*(iv[1]*dx*dy + iv[2]*dx*dz + iv[4]*dy*dz);
    float e = -0.5f * (maha + wLd[kk] + 3.f * LOG2PI_);
    ev[r] = e; emax = fmaxf(emax, e);
  }
  float wgt[4]; float wsum = 0.f;
  #pragma unroll
  for (int r = 0; r < RR_; ++r) { wgt[r] = __expf(ev[r] - emax); wsum += wgt[r]; }
  float invs = 1.f / wsum;
  #pragma unroll
  for (int r = 0; r < RR_; ++r) wgt[r] *= invs;

  float gv = out[O_G + gi];   // written by gateK earlier on this stream
  int k0 = kidx[0], k1 = kidx[1], k2 = kidx[2], k3 = kidx[3];

  // s_child0 = g * sum_r w_r s_parent[k_r]
  const float* spb = sp + (size_t)b * KP_ * CC_;
  float* so = out + (size_t)gi * CC_;
  for (int c = lane; c < CC_; c += 32) {
    float v = wgt[0] * spb[(size_t)k0 * CC_ + c] + wgt[1] * spb[(size_t)k1 * CC_ + c]
            + wgt[2] * spb[(size_t)k2 * CC_ + c] + wgt[3] * spb[(size_t)k3 * CC_ + c];
    so[c] = gv * v;
  }

  // Bmat row: dense write, nonzero only at the 4 top indices
  float* br = out + O_B + (size_t)gi * KP_;
  for (int t = lane; t < KP_ / 4; t += 32) {
    int c0 = t * 4;
    float4 v4;
    int c;
    c = c0 + 0; v4.x = (c==k0)?wgt[0]:(c==k1)?wgt[1]:(c==k2)?wgt[2]:(c==k3)?wgt[3]:0.f;
    c = c0 + 1; v4.y = (c==k0)?wgt[0]:(c==k1)?wgt[1]:(c==k2)?wgt[2]:(c==k3)?wgt[3]:0.f;
    c = c0 + 2; v4.z = (c==k0)?wgt[0]:(c==k1)?wgt[1]:(c==k2)?wgt[2]:(c==k3)?wgt[3]:0.f;
    c = c0 + 3; v4.w = (c==k0)?wgt[0]:(c==k1)?wgt[1]:(c==k2)?wgt[2]:(c==k3)?wgt[3]:0.f;
    *(float4*)(br + c0) = v4;
  }

  if (lane == 0) {
    float mcx = 0.f, mcy = 0.f, mcz = 0.f;
    float s00 = 0.f, s01 = 0.f, s02 = 0.f, s11 = 0.f, s12 = 0.f, s22 = 0.f;
    #pragma unroll
    for (int r = 0; r < RR_; ++r) {
      int kk = b * KP_ + kidx[r];
      float w = wgt[r];
      float mx = mu[(size_t)kk * 3 + 0], my = mu[(size_t)kk * 3 + 1], mz = mu[(size_t)kk * 3 + 2];
      mcx += w * mx; mcy += w * my; mcz += w * mz;
      const float* Sg = Sigma + (size_t)kk * 9;
      s00 += w * (Sg[0] * INVPHI2 + mx * mx);
      s01 += w * (Sg[1] * INVPHI2 + mx * my);
      s02 += w * (Sg[2] * INVPHI2 + mx * mz);
      s11 += w * (Sg[4] * INVPHI2 + my * my);
      s12 += w * (Sg[5] * INVPHI2 + my * mz);
      s22 += w * (Sg[8] * INVPHI2 + mz * mz);
    }
    float rs = wgt[0] + wgt[1] + wgt[2] + wgt[3];
    float f = 2.f - rs;
    s00 -= f * mcx * mcx; s01 -= f * mcx * mcy; s02 -= f * mcx * mcz;
    s11 -= f * mcy * mcy; s12 -= f * mcy * mcz; s22 -= f * mcz * mcz;
    float* mo = out + O_MU + (size_t)gi * 3;
    mo[0] = mcx; mo[1] = mcy; mo[2] = mcz;
    float* So = out + O_SIG + (size_t)gi * 9;
    So[0] = s00 + JITTER_; So[1] = s01;           So[2] = s02;
    So[3] = s01;           So[4] = s11 + JITTER_; So[5] = s12;
    So[6] = s02;           So[7] = s12;           So[8] = s22 + JITTER_;
  }
}

extern "C" void kernel_launch(void* const* d_in, const int* in_sizes, int n_in,
                              void* d_out, int out_size, void* d_ws, size_t ws_size,
                              hipStream_t stream) {
  (void)in_sizes; (void)n_in; (void)out_size; (void)ws_size;
  const float* s_parent    = (const float*)d_in[0];
  const float* mu_parent   = (const float*)d_in[1];
  const float* Sigma_par   = (const float*)d_in[2];
  const float* mask_parent = (const float*)d_in[3];
  const float* xi_noise    = (const float*)d_in[4];
  const float* child_emb   = (const float*)d_in[5];
  const float* geo_w       = (const float*)d_in[6];
  const float* geo_b       = (const float*)d_in[7];
  const float* gate_ln_g   = (const float*)d_in[8];
  const float* gate_ln_b   = (const float*)d_in[9];
  const float* gate_w1     = (const float*)d_in[10];
  const float* gate_b1     = (const float*)d_in[11];
  const float* gate_w2     = (const float*)d_in[12];
  const float* gate_b2     = (const float*)d_in[13];
  // prior MLP inputs d_in[14..19] are dead (BETA == 0)
  float* out = (float*)d_out;
  float* wsf = (float*)d_ws;
  int*   topw = (int*)(wsf + W_TOP);

  prepK<<<(NPAR + 255) / 256, 256, 0, stream>>>(Sigma_par, wsf + W_L, wsf + W_SINV, wsf + W_LD);
  geoK<<<1, 32, 0, stream>>>(child_emb, geo_w, geo_b, wsf + W_XIW);
  gateK<<<NCH / 64, 128, 0, stream>>>(s_parent, child_emb, mask_parent,
                                      gate_ln_g, gate_ln_b, gate_w1, gate_b1,
                                      gate_w2, gate_b2, out);
  topkK<<<NPAR / 8, 256, 0, stream>>>(mu_parent, wsf + W_SINV, wsf + W_LD, topw);
  childK<<<NCH / 8, 256, 0, stream>>>(s_parent, mu_parent, Sigma_par, xi_noise,
                                      wsf + W_L, wsf + W_SINV, wsf + W_LD,
                                      wsf + W_XIW, topw, out);
}